// QProduct_54838142435818
// MI455X (gfx1250) — compile-verified
//
#include <hip/hip_runtime.h>
#include <hip/hip_bf16.h>

// Complex Kronecker chain: out[b,s] = kron(kron((1+1i)*h0, h1), h2), split into
// real/imag fp32 planes of shape (8,128,32768).
//
// Design (MI455X, wave32):
//  - Store-bandwidth bound: 256 MB of output, ~11 us floor at 23.3 TB/s.
//  - One workgroup (256 threads = 8 waves) per (b,s) position; 1024 blocks.
//  - Level-2 (32x32 complex outer product) computed with V_WMMA_F32_16X16X4_F32:
//    only the K=0 slice of A/B is populated, so D = u * v^T. 8 waves cover
//    {real,imag} x {2x2 tiles of 16x16}, 2 chained WMMAs each, results
//    scattered to LDS via the documented 16x16 f32 C/D layout.
//  - Level-3 streamed with float4 non-temporal stores (output > 192MB L2,
//    never re-read -> don't pollute L2).

typedef __attribute__((ext_vector_type(2))) float v2f;
typedef __attribute__((ext_vector_type(8))) float v8f;
typedef __attribute__((ext_vector_type(4))) float f4;

static __device__ __forceinline__ f4 splat4(float x) {
    f4 v = {x, x, x, x};
    return v;
}

__global__ __launch_bounds__(256) void qprod_kron_kernel(
    const float* __restrict__ h0r, const float* __restrict__ h0i,
    const float* __restrict__ h1r, const float* __restrict__ h1i,
    const float* __restrict__ h2r, const float* __restrict__ h2i,
    float* __restrict__ out_r, float* __restrict__ out_i)
{
    __shared__ float tp2r[1024];
    __shared__ float tp2i[1024];

    const int bs   = blockIdx.x;      // (b*128 + s), 1024 blocks
    const int tid  = threadIdx.x;
    const int lane = tid & 31;
    const int wave = tid >> 5;

    const float* H0r = h0r + bs * 32;
    const float* H0i = h0i + bs * 32;
    const float* H1r = h1r + bs * 32;
    const float* H1i = h1i + bs * 32;

    // ---- Phase 1: level-2 = outer((1+1i)*h0, h1) via WMMA outer products ----
    // wave -> (part, tileI, tileJ): part = wave>>2 (0=real, 1=imag)
    const int  part = wave >> 2;
    const int  tI   = (wave >> 1) & 1;
    const int  tJ   = wave & 1;
    const int  li   = lane & 15;
    const bool lo   = lane < 16;

    // tp1 = (1+1i) * h0  =>  tp1_r = h0r - h0i ; tp1_i = h0r + h0i
    const float uhr = H0r[tI * 16 + li];
    const float uhi = H0i[tI * 16 + li];
    const float ur  = lo ? (uhr - uhi) : 0.0f;
    const float ui  = lo ? (uhr + uhi) : 0.0f;
    const float vr  = lo ? H1r[tJ * 16 + li] : 0.0f;
    const float vi  = lo ? H1i[tJ * 16 + li] : 0.0f;

    // A is 16x4 (M x K): VGPR0 holds K=0 (lanes 0-15) / K=2 (lanes 16-31),
    // VGPR1 holds K=1 / K=3. We populate only K=0 => D = u * v^T.
    v2f A0 = {0.0f, 0.0f}, B0 = {0.0f, 0.0f};
    v2f A1 = {0.0f, 0.0f}, B1 = {0.0f, 0.0f};
    if (part == 0) {            // real: u_r (x) v_r - u_i (x) v_i
        A0.x = ur;  B0.x = vr;
        A1.x = -ui; B1.x = vi;
    } else {                    // imag: u_r (x) v_i + u_i (x) v_r
        A0.x = ur;  B0.x = vi;
        A1.x = ui;  B1.x = vr;
    }

    v8f acc = {};
    acc = __builtin_amdgcn_wmma_f32_16x16x4_f32(false, A0, false, B0,
                                                (short)0, acc, false, false);
    acc = __builtin_amdgcn_wmma_f32_16x16x4_f32(false, A1, false, B1,
                                                (short)0, acc, false, false);

    // Scatter 16x16 D tile into LDS. C/D layout: VGPR r holds row M=r
    // (lanes 0-15, N=lane) and row M=r+8 (lanes 16-31, N=lane-16).
    float* dst = part ? tp2i : tp2r;
    const int N     = tJ * 16 + li;
    const int Mbase = tI * 16 + (lo ? 0 : 8);
#pragma unroll
    for (int r = 0; r < 8; ++r) {
        dst[(Mbase + r) * 32 + N] = acc[r];
    }

    __syncthreads();

    // ---- Phase 2: level-3 streaming: out = tp2 (x) h2, NT float4 stores ----
    const float* H2r = h2r + bs * 32;
    const float* H2i = h2i + bs * 32;

    // Thread's fixed k-group of 4 (tid % 8 is invariant across iterations).
    const f4 h2r4 = *(const f4*)(H2r + (tid & 7) * 4);
    const f4 h2i4 = *(const f4*)(H2i + (tid & 7) * 4);

    float* Or = out_r + (size_t)bs * 32768;
    float* Oi = out_i + (size_t)bs * 32768;

#pragma unroll 4
    for (int t = 0; t < 32; ++t) {
        const int idx4 = t * 256 + tid;   // float4 index, 0..8191
        const int D    = idx4 >> 3;       // level-2 element index, 0..1023
        const float tr = tp2r[D];
        const float ti = tp2i[D];
        const f4 r4 = splat4(tr) * h2r4 - splat4(ti) * h2i4;
        const f4 i4 = splat4(tr) * h2i4 + splat4(ti) * h2r4;
        __builtin_nontemporal_store(r4, (f4*)(Or + (size_t)idx4 * 4));
        __builtin_nontemporal_store(i4, (f4*)(Oi + (size_t)idx4 * 4));
    }
}

extern "C" void kernel_launch(void* const* d_in, const int* in_sizes, int n_in,
                              void* d_out, int out_size, void* d_ws, size_t ws_size,
                              hipStream_t stream) {
    const float* h0r = (const float*)d_in[0];
    const float* h0i = (const float*)d_in[1];
    const float* h1r = (const float*)d_in[2];
    const float* h1i = (const float*)d_in[3];
    const float* h2r = (const float*)d_in[4];
    const float* h2i = (const float*)d_in[5];

    float* out   = (float*)d_out;
    float* out_r = out;                                   // (8,128,32768) real
    float* out_i = out + (size_t)8 * 128 * 32768;         // (8,128,32768) imag

    qprod_kron_kernel<<<dim3(1024), dim3(256), 0, stream>>>(
        h0r, h0i, h1r, h1i, h2r, h2i, out_r, out_i);
}